// NonMaxSuppression_2662879724404
// MI455X (gfx1250) — compile-verified
//
#include <hip/hip_runtime.h>
#include <hip/hip_bf16.h>
#include <stdint.h>

// Problem constants (match reference)
#define B_       16
#define N_       25200
#define C_       80
#define MAXDET   100
#define IOU_THR  0.5f
#define CONF_THR 0.5f

// ---- Phase 1 (class max/argmax) tiling ----
#define TILE            96                       // boxes per tile / threads per block (3 wave32)
#define TILES_PER_BATCH ((N_ + TILE - 1) / TILE) // 263
#define ROWP            85                       // 80 data dwords + 5 TDM pad dwords per row

// ---- Phase 2 (NMS) ----
#define NMS_THREADS 1024
#define PER 25  // ceil(25200/1024): scores+areas per thread kept in VGPRs

#if defined(__has_builtin)
#if __has_builtin(__builtin_amdgcn_tensor_load_to_lds) && \
    __has_builtin(__builtin_amdgcn_s_wait_tensorcnt)
#define USE_TDM 1
#endif
#endif

typedef unsigned int u32x4 __attribute__((ext_vector_type(4)));
typedef int          i32x4 __attribute__((ext_vector_type(4)));
typedef int          i32x8 __attribute__((ext_vector_type(8)));

// ---------------------------------------------------------------------------
// Kernel A: conf[b,n] = max_c classes[b,n,c]; cls[b,n] = first argmax.
// TDM path: wave 0 issues a descriptor-driven DMA of the tile (nvalid*80 f32,
// 1-D) into LDS with 1-dword padding every 16 dwords -> row stride 85 dwords
// (coprime with 64 banks => conflict-free per-row scans). TENSORcnt-tracked.
// ---------------------------------------------------------------------------
__global__ __launch_bounds__(TILE) void conf_argmax_kernel(
    const float* __restrict__ classes,
    float* __restrict__ conf,
    int* __restrict__ cls) {
  const int tile   = blockIdx.x % TILES_PER_BATCH;
  const int b      = blockIdx.x / TILES_PER_BATCH;
  const int box0   = tile * TILE;
  const int nvalid = (N_ - box0 < TILE) ? (N_ - box0) : TILE;
  const int t      = threadIdx.x;

#ifdef USE_TDM
  __shared__ float cbuf[TILE * ROWP];  // 8160 dwords = 31.9 KB
  if (t < 32) {  // wave 0 only: TDM ignores EXEC; one issue per tile
    const uint64_t ga =
        (uint64_t)(uintptr_t)(classes + ((uint64_t)b * N_ + (uint64_t)box0) * C_);
    const uint32_t ldsa  = (uint32_t)(uintptr_t)(&cbuf[0]);
    const uint32_t elems = (uint32_t)nvalid * (uint32_t)C_;  // <= 7680 (< 2^16)

    // D# group 0: count=1 | lds_addr | global_addr[56:0] | type=2
    u32x4 g0;
    g0.x = 1u;
    g0.y = ldsa;
    g0.z = (uint32_t)ga;
    g0.w = (uint32_t)((ga >> 32) & 0x1FFFFFFull) | (2u << 30);

    // D# group 1: data_size=4B(2), pad_enable, pad_interval=16dw(3),
    // pad_amount=1dw(0); tensor_dim0 = tile_dim0 = elems (1-D copy),
    // tensor_dim1 = 1, tile_dim1 = 1, stride0 = elems.
    i32x8 g1;
    g1[0] = (int)((2u << 16) | (1u << 20) | (3u << 22));
    g1[1] = (int)((elems & 0xFFFFu) << 16);                  // tensor_dim0 lo
    g1[2] = (int)(((elems >> 16) & 0xFFFFu) | (1u << 16));   // dim0 hi | tensor_dim1=1
    g1[3] = (int)((elems & 0xFFFFu) << 16);                  // dim1 hi=0 | tile_dim0
    g1[4] = 1;                                               // tile_dim1=1, tile_dim2=0
    g1[5] = (int)elems;                                      // tensor_dim0_stride lo
    g1[6] = 0;
    g1[7] = 0;
    i32x4 gz4 = {0, 0, 0, 0};
    i32x8 gz8 = {0, 0, 0, 0, 0, 0, 0, 0};

    // clang-23 / amdgpu-toolchain arity (probe-confirmed 6 args):
    // (u32x4 g0, i32x8 g1, i32x4 g2, i32x4 g3, i32x8, i32 cpol)
    __builtin_amdgcn_tensor_load_to_lds(g0, g1, gz4, gz4, gz8, 0);
    __builtin_amdgcn_s_wait_tensorcnt(0);
  }
  __syncthreads();

  if (t < nvalid) {
    const float* row = &cbuf[t * ROWP];
    float best = row[0];
    int   bi   = 0;
#pragma unroll
    for (int k = 1; k < C_; ++k) {
      float v = row[k + (k >> 4)];  // skip pad dword every 16
      if (v > best) { best = v; bi = k; }  // strict > == first-max tie-break
    }
    conf[(size_t)b * N_ + box0 + t] = best;
    cls [(size_t)b * N_ + box0 + t] = bi;
  }
#else
  // Fallback: direct streaming float4 loads (still HBM-bound optimal).
  if (t < nvalid) {
    const float4* r4 =
        (const float4*)(classes + ((uint64_t)b * N_ + (uint64_t)(box0 + t)) * C_);
    float best = -INFINITY;
    int   bi   = 0;
#pragma unroll
    for (int q = 0; q < C_ / 4; ++q) {
      float4 v = r4[q];
      int k = 4 * q;
      if (v.x > best) { best = v.x; bi = k; }
      if (v.y > best) { best = v.y; bi = k + 1; }
      if (v.z > best) { best = v.z; bi = k + 2; }
      if (v.w > best) { best = v.w; bi = k + 3; }
    }
    conf[(size_t)b * N_ + box0 + t] = best;
    cls [(size_t)b * N_ + box0 + t] = bi;
  }
#endif
}

// ---------------------------------------------------------------------------
// Kernel B: exact greedy NMS. One 1024-thread block (32 wave32) per batch.
// Scores + areas live in VGPRs (25/thread). Argmax = per-thread scan ->
// wave32 shfl_xor butterfly -> 32-entry LDS reduction by wave 0. Winner box
// broadcast-read from L2 (same address all lanes). Boxes re-streamed from L2
// (403 KB/batch, resident in 192 MB L2) each suppression pass.
// ---------------------------------------------------------------------------
__global__ __launch_bounds__(NMS_THREADS) void nms_kernel(
    const float* __restrict__ boxes, const float* __restrict__ conf,
    const int* __restrict__ cls, float* __restrict__ out_boxes,
    float* __restrict__ out_conf, float* __restrict__ out_cls,
    float* __restrict__ out_num) {
  const int    b  = blockIdx.x;
  const int    t  = threadIdx.x;
  const size_t bn = (size_t)b * N_;
  const float4* bx4 = (const float4*)boxes + bn;

  float s[PER];
  float area[PER];
#pragma unroll
  for (int k = 0; k < PER; ++k) {
    int n = t + k * NMS_THREADS;
    if (n < N_) {
      float c = conf[bn + n];
      s[k] = (c > CONF_THR) ? c : -INFINITY;
      float4 bb = bx4[n];
      area[k] = (bb.z - bb.x) * (bb.w - bb.y);
    } else {
      s[k]    = -INFINITY;
      area[k] = 0.0f;
    }
  }

  __shared__ float sval[NMS_THREADS / 32];
  __shared__ int   sidx[NMS_THREADS / 32];
  __shared__ float bval;
  __shared__ int   bidx;

  int ndet = 0;
  for (int it = 0; it < MAXDET; ++it) {
    // --- global argmax with smallest-index tie-break (matches jnp.argmax) ---
    float lv = -INFINITY;
    int   li = 0x7FFFFFFF;
#pragma unroll
    for (int k = 0; k < PER; ++k) {
      int   n = t + k * NMS_THREADS;
      float v = s[k];
      if (v > lv || (v == lv && n < li)) { lv = v; li = n; }
    }
#pragma unroll
    for (int off = 16; off > 0; off >>= 1) {
      float ov = __shfl_xor(lv, off, 32);
      int   oi = __shfl_xor(li, off, 32);
      if (ov > lv || (ov == lv && oi < li)) { lv = ov; li = oi; }
    }
    if ((t & 31) == 0) { sval[t >> 5] = lv; sidx[t >> 5] = li; }
    __syncthreads();
    if (t < 32) {
      lv = sval[t];
      li = sidx[t];
#pragma unroll
      for (int off = 16; off > 0; off >>= 1) {
        float ov = __shfl_xor(lv, off, 32);
        int   oi = __shfl_xor(li, off, 32);
        if (ov > lv || (ov == lv && oi < li)) { lv = ov; li = oi; }
      }
      if (t == 0) { bval = lv; bidx = li; }
    }
    __syncthreads();

    const int   wi = bidx;           // == 0 when everything is -inf (tie-break)
    const float wv = bval;
    const bool  ok = wv > -INFINITY;
    const float4 wb = bx4[wi];       // uniform address -> broadcast from L2

    if (t == 0) {
      ((float4*)out_boxes)[(size_t)b * MAXDET + it] = wb;       // boxes[idx]
      out_conf[b * MAXDET + it] = conf[bn + wi];                // unmasked conf
      out_cls [b * MAXDET + it] = (float)cls[bn + wi];          // per-box argmax
      ndet += ok ? 1 : 0;
    }

    if (ok) {  // block-uniform; when !ok everything is already -inf (no-op)
      const float wy1 = wb.x, wx1 = wb.y, wy2 = wb.z, wx2 = wb.w;
      const float wa  = (wy2 - wy1) * (wx2 - wx1);
#pragma unroll
      for (int k = 0; k < PER; ++k) {
        int n = t + k * NMS_THREADS;
        if (n < N_) {
          float4 bb  = bx4[n];
          float yy1  = fmaxf(wy1, bb.x);
          float xx1  = fmaxf(wx1, bb.y);
          float yy2  = fminf(wy2, bb.z);
          float xx2  = fminf(wx2, bb.w);
          float inter = fmaxf(yy2 - yy1, 0.0f) * fmaxf(xx2 - xx1, 0.0f);
          float iou   = inter / fmaxf(wa + area[k] - inter, 1e-9f);
          if (iou > IOU_THR || n == wi) s[k] = -INFINITY;
        }
      }
    }
  }
  if (t == 0) out_num[b] = (float)ndet;
}

// ---------------------------------------------------------------------------
extern "C" void kernel_launch(void* const* d_in, const int* in_sizes, int n_in,
                              void* d_out, int out_size, void* d_ws, size_t ws_size,
                              hipStream_t stream) {
  const float* boxes   = (const float*)d_in[0];  // [B, N, 4] f32
  const float* classes = (const float*)d_in[1];  // [B, N, C] f32

  // workspace: conf [B*N] f32, then cls [B*N] i32 (~3.1 MB total)
  float* conf = (float*)d_ws;
  int*   cls  = (int*)((char*)d_ws + (size_t)B_ * N_ * sizeof(float));

  // outputs flat in return order: boxes[6400] | conf[1600] | classes[1600] | num[16]
  float* out_boxes = (float*)d_out;
  float* out_conf  = out_boxes + (size_t)B_ * MAXDET * 4;
  float* out_cls   = out_conf  + (size_t)B_ * MAXDET;
  float* out_num   = out_cls   + (size_t)B_ * MAXDET;

  conf_argmax_kernel<<<dim3(B_ * TILES_PER_BATCH), dim3(TILE), 0, stream>>>(
      classes, conf, cls);
  nms_kernel<<<dim3(B_), dim3(NMS_THREADS), 0, stream>>>(
      boxes, conf, cls, out_boxes, out_conf, out_cls, out_num);
}